// HierarchicalGraphReasoner_14860586844393
// MI455X (gfx1250) — compile-verified
//
#include <hip/hip_runtime.h>
#include <math.h>

#define HIDN 512
#define HEADS 4
#define DHD 128
#define NLAY 2
#define WINR 5
#define BSZ 16
#define SEQ 1024
#define NNODE 1026
#define NROWS (BSZ * NNODE)   // 16416
#define SLOPE 0.2f
#define LNEPS 1e-5f
#define FFH 2048

typedef __attribute__((ext_vector_type(2))) float v2f;
typedef __attribute__((ext_vector_type(8))) float v8f;

// Row mapping: float offset of row r is base + ((r+off)/rpb)*bstride + ((r+off)%rpb)*rstride
struct RowMap {
  long base;
  long rpb;
  long bstride;
  long rstride;
  long off;
};

__device__ __forceinline__ long map_row(const RowMap m, long r) {
  long rr = r + m.off;
  return m.base + (rr / m.rpb) * m.bstride + (rr % m.rpb) * m.rstride;
}

// ---------------------------------------------------------------------------
// Tiled f32 WMMA GEMM:  C = epilogue(A * B [+ bias] [+ res])
// Block tile 64x64, K tile 32, 256 threads = 8 waves (4 M-subtiles x 2 N-halves).
// Double-buffered LDS staged with GLOBAL_LOAD_ASYNC_TO_LDS_B128/B32 (ASYNCcnt),
// compute via V_WMMA_F32_16X16X4_F32.
// ---------------------------------------------------------------------------
template <int GELU, int TRANSB, int HASBIAS, int HASRES>
__global__ __launch_bounds__(256) void gemm_wmma_f32(
    const float* __restrict__ A, const float* __restrict__ Bm,
    const float* __restrict__ bias, const float* __restrict__ Res,
    float* __restrict__ C, int M, int K, int Nn,
    RowMap amap, RowMap cmap, RowMap rmap) {
  // Row pads chosen so every async B128 target and every v2f A read is 16B/8B
  // aligned (stride 36/68 floats, both = 0 mod 4) and LDS-bank conflict free.
  __shared__ float As[2][64][36] __attribute__((aligned(16)));
  __shared__ float Bs[2][32][68] __attribute__((aligned(16)));
  const int t = threadIdx.x;
  const int lane = t & 31;
  const int w = t >> 5;   // wave 0..7
  const int wm = w & 3;   // 16-row M subtile
  const int wn = w >> 2;  // 32-col N half
  const long blockM = (long)blockIdx.y * 64;
  const long blockN = (long)blockIdx.x * 64;
  const int mrow = lane & 15;
  const int klane = (lane >> 4) * 2;  // lanes 16..31 hold K+2,K+3 (ISA f32 A/B layout)

  const unsigned long long abase = (unsigned long long)A;
  const unsigned long long bbase = (unsigned long long)Bm;

  // Issue async memory->LDS copies for the K-tile starting at k0 into buffer `buf`.
  auto issue_tile = [&](int buf, int k0) {
    // A tile: 64x32 floats as 512 x 16B segments, 2 per thread
#pragma unroll
    for (int i = 0; i < 2; ++i) {
      int s = t + i * 256;
      int r = s >> 3;
      int c = (s & 7) * 4;
      long gm = blockM + r;
      if (gm < M) {
        unsigned lds = (unsigned)(size_t)&As[buf][r][c];
        unsigned goff = (unsigned)((map_row(amap, gm) + (k0 + c)) * 4);
        asm volatile("global_load_async_to_lds_b128 %0, %1, %2"
                     :: "v"(lds), "v"(goff), "s"(abase) : "memory");
      }
    }
    if (TRANSB) {
      // B[k][n] = Bm[n*K + k]: global contiguous along K, LDS contiguous along N
      // -> element-wise B32 async copies (transpose happens via LDS placement)
#pragma unroll
      for (int i = 0; i < 8; ++i) {
        int e = t + i * 256;
        int kk = e & 31, nn = e >> 5;
        long gn = blockN + nn;
        if (gn < Nn) {
          unsigned lds = (unsigned)(size_t)&Bs[buf][kk][nn];
          unsigned goff = (unsigned)((gn * (long)K + (k0 + kk)) * 4);
          asm volatile("global_load_async_to_lds_b32 %0, %1, %2"
                       :: "v"(lds), "v"(goff), "s"(bbase) : "memory");
        }
      }
    } else {
      // B tile: 32x64 floats as 512 x 16B segments, 2 per thread
#pragma unroll
      for (int i = 0; i < 2; ++i) {
        int s = t + i * 256;
        int kk = s >> 4;
        int nn = (s & 15) * 4;
        long gn = blockN + nn;
        if (gn < Nn) {
          unsigned lds = (unsigned)(size_t)&Bs[buf][kk][nn];
          unsigned goff = (unsigned)(((long)(k0 + kk) * Nn + gn) * 4);
          asm volatile("global_load_async_to_lds_b128 %0, %1, %2"
                       :: "v"(lds), "v"(goff), "s"(bbase) : "memory");
        }
      }
    }
  };

  v8f acc0 = {0.f, 0.f, 0.f, 0.f, 0.f, 0.f, 0.f, 0.f};
  v8f acc1 = {0.f, 0.f, 0.f, 0.f, 0.f, 0.f, 0.f, 0.f};

  int buf = 0;
  issue_tile(0, 0);
  asm volatile("s_wait_asynccnt 0" ::: "memory");
  __syncthreads();

  for (int k0 = 0; k0 < K; k0 += 32) {
    const int nxt = k0 + 32;
    if (nxt < K) issue_tile(buf ^ 1, nxt);  // overlap next-tile DMA with WMMAs
#pragma unroll
    for (int kk = 0; kk < 32; kk += 4) {
      const int kb = kk + klane;
      v2f a = *reinterpret_cast<const v2f*>(&As[buf][wm * 16 + mrow][kb]);
      v2f b0;
      b0.x = Bs[buf][kb][wn * 32 + mrow];
      b0.y = Bs[buf][kb + 1][wn * 32 + mrow];
      v2f b1;
      b1.x = Bs[buf][kb][wn * 32 + 16 + mrow];
      b1.y = Bs[buf][kb + 1][wn * 32 + 16 + mrow];
      acc0 = __builtin_amdgcn_wmma_f32_16x16x4_f32(false, a, false, b0,
                                                   (short)0, acc0, false, false);
      acc1 = __builtin_amdgcn_wmma_f32_16x16x4_f32(false, a, false, b1,
                                                   (short)0, acc1, false, false);
    }
    asm volatile("s_wait_asynccnt 0" ::: "memory");  // my async writes to buf^1 done
    __syncthreads();                                  // everyone done reading buf
    buf ^= 1;
  }

  // Epilogue: D layout (16x16 f32): VGPR r -> M=r (lanes 0-15) / M=8+r (16-31), N=lane&15
  const long mb = blockM + wm * 16 + (lane >> 4) * 8;
#pragma unroll
  for (int r = 0; r < 8; ++r) {
    long gm = mb + r;
    if (gm >= M) continue;
    long crow = map_row(cmap, gm);
    long rrow = HASRES ? map_row(rmap, gm) : 0;
#pragma unroll
    for (int nt = 0; nt < 2; ++nt) {
      long gn = blockN + wn * 32 + nt * 16 + mrow;
      if (gn >= Nn) continue;
      float v = (nt == 0) ? acc0[r] : acc1[r];
      if (HASBIAS) v += bias[gn];
      if (GELU) v = 0.5f * v * (1.f + erff(v * 0.70710678118654752440f));
      if (HASRES) v += Res[rrow + gn];
      C[crow + gn] = v;
    }
  }
}

// ---------------------------------------------------------------------------
// nodes[b] = concat(global, tokens, local)
// ---------------------------------------------------------------------------
__global__ void build_nodes_kernel(const float* __restrict__ g,
                                   const float* __restrict__ tok,
                                   const float* __restrict__ l,
                                   float* __restrict__ nodes) {
  long idx = (long)blockIdx.x * 256 + threadIdx.x;
  long row = idx >> 9;
  int d = (int)(idx & 511);
  int b = (int)(row / NNODE);
  int pos = (int)(row % NNODE);
  float v;
  if (pos == 0) v = g[b * HIDN + d];
  else if (pos == NNODE - 1) v = l[b * HIDN + d];
  else v = tok[((long)b * SEQ + pos - 1) * HIDN + d];
  nodes[idx] = v;
}

// ---------------------------------------------------------------------------
// sq[row,h] = P[row, h*128 : h*128+128] . a[h, :128]; sk with a[h, 128:]
// ---------------------------------------------------------------------------
__global__ void sqsk_kernel(const float* __restrict__ P,
                            const float* __restrict__ a,
                            float* __restrict__ sq, float* __restrict__ sk) {
  const long row = blockIdx.x;
  const int t = threadIdx.x;
  const int h = t >> 7, d = t & 127;
  __shared__ float r1[512], r2[512];
  float p = P[row * HIDN + t];
  r1[t] = p * a[h * 2 * DHD + d];
  r2[t] = p * a[h * 2 * DHD + DHD + d];
  __syncthreads();
  for (int s = 64; s > 0; s >>= 1) {
    if (d < s) { r1[t] += r1[t + s]; r2[t] += r2[t + s]; }
    __syncthreads();
  }
  if (d == 0) {
    sq[row * HEADS + h] = r1[t];
    sk[row * HEADS + h] = r2[t];
  }
}

// ---------------------------------------------------------------------------
// Sparse token attention: <=13 keys per query (global + 11-window + optional local)
// ---------------------------------------------------------------------------
__global__ void token_attn_kernel(const float* __restrict__ P,
                                  const float* __restrict__ sq,
                                  const float* __restrict__ sk,
                                  const float* __restrict__ amask,
                                  const int* __restrict__ last,
                                  float* __restrict__ attn) {
  const int b = blockIdx.x >> 10;
  const int i = blockIdx.x & 1023;
  const int t = threadIdx.x;  // 512
  const long rowq = (long)b * NNODE + 1 + i;
  __shared__ int nk;
  __shared__ int keys[13];
  __shared__ float wgt[13 * HEADS];
  const float qm = amask[b * SEQ + i];
  if (t == 0) {
    int cnt = 0;
    keys[cnt++] = 0;  // global node, pad column is always 1
    int lo = (i - WINR < 0) ? 0 : i - WINR;
    int hi = (i + WINR > SEQ - 1) ? SEQ - 1 : i + WINR;
    for (int tt = lo; tt <= hi; ++tt)
      if (amask[b * SEQ + tt] > 0.f) keys[cnt++] = tt + 1;
    if (i >= last[b]) keys[cnt++] = NNODE - 1;  // local node
    nk = cnt;
  }
  __syncthreads();
  if (t < HEADS) {
    const int h = t;
    const float sqq = sq[rowq * HEADS + h];
    float sc[13];
    float mx = -1e30f;
    for (int k = 0; k < nk; ++k) {
      float s = sqq + sk[((long)b * NNODE + keys[k]) * HEADS + h];
      s = (s >= 0.f) ? s : SLOPE * s;
      sc[k] = s;
      mx = fmaxf(mx, s);
    }
    float sum = 0.f;
    for (int k = 0; k < nk; ++k) {
      float e = expf(sc[k] - mx);
      wgt[k * HEADS + h] = e;
      sum += e;
    }
    float inv = 1.f / sum;
    for (int k = 0; k < nk; ++k) wgt[k * HEADS + h] *= inv;
  }
  __syncthreads();
  const int h = t >> 7;
  float acc = 0.f;
  for (int k = 0; k < nk; ++k)
    acc += wgt[k * HEADS + h] * P[((long)b * NNODE + keys[k]) * HIDN + t];
  attn[rowq * HIDN + t] = (qm > 0.f) ? acc : 0.f;
}

// ---------------------------------------------------------------------------
// Dense attention for global (all 1026 keys) and local (position mask) queries
// ---------------------------------------------------------------------------
__global__ void gl_attn_kernel(const float* __restrict__ P,
                               const float* __restrict__ sq,
                               const float* __restrict__ sk,
                               const int* __restrict__ last,
                               float* __restrict__ attn) {
  const int b = blockIdx.x & 15;
  const bool isLocal = blockIdx.x >= 16;
  const int qpos = isLocal ? (NNODE - 1) : 0;
  const long rowq = (long)b * NNODE + qpos;
  const int t = threadIdx.x;  // 256
  const int lst = last[b];
  __shared__ float red[HEADS * 256];
  float sqq[HEADS];
  for (int h = 0; h < HEADS; ++h) sqq[h] = sq[rowq * HEADS + h];

  float mx[HEADS] = {-1e30f, -1e30f, -1e30f, -1e30f};
  for (int j = t; j < NNODE; j += 256) {
    bool ok = !isLocal || (j == 0) || (j == NNODE - 1) || (j >= lst + 1 && j <= SEQ);
    if (ok) {
      for (int h = 0; h < HEADS; ++h) {
        float s = sqq[h] + sk[((long)b * NNODE + j) * HEADS + h];
        s = (s >= 0.f) ? s : SLOPE * s;
        mx[h] = fmaxf(mx[h], s);
      }
    }
  }
  for (int h = 0; h < HEADS; ++h) red[h * 256 + t] = mx[h];
  __syncthreads();
  for (int s = 128; s > 0; s >>= 1) {
    if (t < s)
      for (int h = 0; h < HEADS; ++h)
        red[h * 256 + t] = fmaxf(red[h * 256 + t], red[h * 256 + t + s]);
    __syncthreads();
  }
  for (int h = 0; h < HEADS; ++h) mx[h] = red[h * 256];
  __syncthreads();

  float sm[HEADS] = {0.f, 0.f, 0.f, 0.f};
  for (int j = t; j < NNODE; j += 256) {
    bool ok = !isLocal || (j == 0) || (j == NNODE - 1) || (j >= lst + 1 && j <= SEQ);
    if (ok) {
      for (int h = 0; h < HEADS; ++h) {
        float s = sqq[h] + sk[((long)b * NNODE + j) * HEADS + h];
        s = (s >= 0.f) ? s : SLOPE * s;
        sm[h] += expf(s - mx[h]);
      }
    }
  }
  for (int h = 0; h < HEADS; ++h) red[h * 256 + t] = sm[h];
  __syncthreads();
  for (int s = 128; s > 0; s >>= 1) {
    if (t < s)
      for (int h = 0; h < HEADS; ++h) red[h * 256 + t] += red[h * 256 + t + s];
    __syncthreads();
  }
  for (int h = 0; h < HEADS; ++h) sm[h] = red[h * 256];

  for (int half = 0; half < 2; ++half) {
    const int d = t + half * 256;
    const int h = d >> 7;
    float acc = 0.f;
    for (int j = 0; j < NNODE; ++j) {
      bool ok = !isLocal || (j == 0) || (j == NNODE - 1) || (j >= lst + 1 && j <= SEQ);
      if (ok) {
        float s = sqq[h] + sk[((long)b * NNODE + j) * HEADS + h];
        s = (s >= 0.f) ? s : SLOPE * s;
        acc += (expf(s - mx[h]) / sm[h]) * P[((long)b * NNODE + j) * HIDN + d];
      }
    }
    attn[rowq * HIDN + d] = acc;
  }
}

// ---------------------------------------------------------------------------
// nodes[row] = LayerNorm(nodes[row] + elu(attn[row]))
// ---------------------------------------------------------------------------
__global__ void res_ln_kernel(float* __restrict__ nodes,
                              const float* __restrict__ attn,
                              const float* __restrict__ gamma,
                              const float* __restrict__ beta) {
  const long row = blockIdx.x;
  const int t = threadIdx.x;  // 256
  float x0 = nodes[row * HIDN + t];
  float x1 = nodes[row * HIDN + t + 256];
  float a0 = attn[row * HIDN + t];
  float a1 = attn[row * HIDN + t + 256];
  a0 = (a0 > 0.f) ? a0 : expf(a0) - 1.f;
  a1 = (a1 > 0.f) ? a1 : expf(a1) - 1.f;
  x0 += a0;
  x1 += a1;
  __shared__ float rs[256], rq[256];
  rs[t] = x0 + x1;
  rq[t] = x0 * x0 + x1 * x1;
  __syncthreads();
  for (int s = 128; s > 0; s >>= 1) {
    if (t < s) { rs[t] += rs[t + s]; rq[t] += rq[t + s]; }
    __syncthreads();
  }
  const float mean = rs[0] * (1.f / HIDN);
  float var = rq[0] * (1.f / HIDN) - mean * mean;
  var = (var > 0.f) ? var : 0.f;
  const float rstd = rsqrtf(var + LNEPS);
  nodes[row * HIDN + t] = (x0 - mean) * rstd * gamma[t] + beta[t];
  nodes[row * HIDN + t + 256] = (x1 - mean) * rstd * gamma[t + 256] + beta[t + 256];
}

// ---------------------------------------------------------------------------
// Final LN: reads post-FFN node buffer, writes d_out in (global, tokens, local) order
// ---------------------------------------------------------------------------
__global__ void final_ln_kernel(const float* __restrict__ post,
                                const float* __restrict__ og,
                                const float* __restrict__ ob,
                                float* __restrict__ out) {
  const long row = blockIdx.x;
  const int t = threadIdx.x;  // 256
  const int b = (int)(row / NNODE);
  const int pos = (int)(row % NNODE);
  int idx;
  long obase;
  if (pos == 0) { idx = 0; obase = (long)b * HIDN; }
  else if (pos == NNODE - 1) {
    idx = 2;
    obase = (long)BSZ * HIDN + (long)BSZ * SEQ * HIDN + (long)b * HIDN;
  } else {
    idx = 1;
    obase = (long)BSZ * HIDN + ((long)b * SEQ + pos - 1) * HIDN;
  }
  float x0 = post[row * HIDN + t];
  float x1 = post[row * HIDN + t + 256];
  __shared__ float rs[256], rq[256];
  rs[t] = x0 + x1;
  rq[t] = x0 * x0 + x1 * x1;
  __syncthreads();
  for (int s = 128; s > 0; s >>= 1) {
    if (t < s) { rs[t] += rs[t + s]; rq[t] += rq[t + s]; }
    __syncthreads();
  }
  const float mean = rs[0] * (1.f / HIDN);
  float var = rq[0] * (1.f / HIDN) - mean * mean;
  var = (var > 0.f) ? var : 0.f;
  const float rstd = rsqrtf(var + LNEPS);
  const float* g = og + (long)idx * HIDN;
  const float* bb = ob + (long)idx * HIDN;
  out[obase + t] = (x0 - mean) * rstd * g[t] + bb[t];
  out[obase + t + 256] = (x1 - mean) * rstd * g[t + 256] + bb[t + 256];
}

// ---------------------------------------------------------------------------
extern "C" void kernel_launch(void* const* d_in, const int* in_sizes, int n_in,
                              void* d_out, int out_size, void* d_ws, size_t ws_size,
                              hipStream_t stream) {
  (void)in_sizes; (void)n_in; (void)out_size; (void)ws_size;
  const float* g_emb = (const float*)d_in[0];
  const float* tok   = (const float*)d_in[1];
  const float* l_emb = (const float*)d_in[2];
  const float* amask = (const float*)d_in[3];
  const int*   last  = (const int*)d_in[4];
  const float* gatW  = (const float*)d_in[5];
  const float* gatA  = (const float*)d_in[6];
  const float* lng   = (const float*)d_in[7];
  const float* lnb   = (const float*)d_in[8];
  const float* w1    = (const float*)d_in[9];
  const float* b1    = (const float*)d_in[10];
  const float* w2    = (const float*)d_in[11];
  const float* b2    = (const float*)d_in[12];
  const float* olg   = (const float*)d_in[13];
  const float* olb   = (const float*)d_in[14];
  float* out = (float*)d_out;

  float* ws    = (float*)d_ws;
  float* nodes = ws;                                 // NROWS*512
  float* P     = nodes + (size_t)NROWS * HIDN;       // NROWS*512
  float* post  = P + (size_t)NROWS * HIDN;           // NROWS*512 (attn during layers)
  float* mid   = post + (size_t)NROWS * HIDN;        // 4096*2048
  float* sq    = mid + (size_t)4096 * FFH;           // NROWS*4
  float* sk    = sq + (size_t)NROWS * HEADS;         // NROWS*4
  float* attn  = post;                               // reuse

  const RowMap mapC512{0, (long)1 << 40, 0, HIDN, 0};   // contiguous ld=512
  const RowMap mapMid{0, (long)1 << 40, 0, FFH, 0};     // contiguous ld=2048

  // 1) assemble node buffer
  build_nodes_kernel<<<(NROWS * HIDN) / 256, 256, 0, stream>>>(g_emb, tok, l_emb, nodes);

  // 2) GAT layers
  for (int k = 0; k < NLAY; ++k) {
    // P = nodes @ W_k^T  (single projection serves Q, K and V for g/h/l)
    gemm_wmma_f32<0, 1, 0, 0>
        <<<dim3(HIDN / 64, (NROWS + 63) / 64), 256, 0, stream>>>(
            nodes, gatW + (size_t)k * HIDN * HIDN, nullptr, nullptr, P,
            NROWS, HIDN, HIDN, mapC512, mapC512, mapC512);
    sqsk_kernel<<<NROWS, 512, 0, stream>>>(P, gatA + (size_t)k * HEADS * 2 * DHD, sq, sk);
    token_attn_kernel<<<BSZ * SEQ, 512, 0, stream>>>(P, sq, sk, amask, last, attn);
    gl_attn_kernel<<<2 * BSZ, 256, 0, stream>>>(P, sq, sk, last, attn);
    res_ln_kernel<<<NROWS, 256, 0, stream>>>(nodes, attn, lng + (size_t)k * HIDN,
                                             lnb + (size_t)k * HIDN);
  }

  // 3) FFNs (tokens in 4 row-chunks of 4096 to bound the 2048-wide intermediate)
  for (int c = 0; c < 4; ++c) {
    const RowMap tokMap{(long)HIDN, SEQ, (long)NNODE * HIDN, HIDN, (long)c * 4096};
    gemm_wmma_f32<1, 0, 1, 0><<<dim3(FFH / 64, 4096 / 64), 256, 0, stream>>>(
        nodes, w1 + (size_t)1 * HIDN * FFH, b1 + FFH, nullptr, mid,
        4096, HIDN, FFH, tokMap, mapMid, mapMid);
    gemm_wmma_f32<0, 0, 1, 1><<<dim3(HIDN / 64, 4096 / 64), 256, 0, stream>>>(
        mid, w2 + (size_t)1 * FFH * HIDN, b2 + HIDN, nodes, post,
        4096, FFH, HIDN, mapMid, tokMap, tokMap);
  }
  {
    const RowMap gMap{0, 1, (long)NNODE * HIDN, HIDN, 0};
    gemm_wmma_f32<1, 0, 1, 0><<<dim3(FFH / 64, 1), 256, 0, stream>>>(
        nodes, w1, b1, nullptr, mid, BSZ, HIDN, FFH, gMap, mapMid, mapMid);
    gemm_wmma_f32<0, 0, 1, 1><<<dim3(HIDN / 64, 1), 256, 0, stream>>>(
        mid, w2, b2, nodes, post, BSZ, FFH, HIDN, mapMid, gMap, gMap);
  }
  {
    const RowMap lMap{(long)(NNODE - 1) * HIDN, 1, (long)NNODE * HIDN, HIDN, 0};
    gemm_wmma_f32<1, 0, 1, 0><<<dim3(FFH / 64, 1), 256, 0, stream>>>(
        nodes, w1 + (size_t)2 * HIDN * FFH, b1 + 2 * FFH, nullptr, mid,
        BSZ, HIDN, FFH, lMap, mapMid, mapMid);
    gemm_wmma_f32<0, 0, 1, 1><<<dim3(HIDN / 64, 1), 256, 0, stream>>>(
        mid, w2 + (size_t)2 * FFH * HIDN, b2 + 2 * HIDN, nodes, post,
        BSZ, FFH, HIDN, mapMid, lMap, lMap);
  }

  // 4) final LayerNorms into d_out (global | tokens | local)
  final_ln_kernel<<<NROWS, 256, 0, stream>>>(post, olg, olb, out);
}